// PureMamba_80221399155272
// MI455X (gfx1250) — compile-verified
//
#include <hip/hip_runtime.h>
#include <hip/hip_bf16.h>
#include <math.h>

// ---- problem constants (from reference) ----
#define BB   8
#define LL   2048
#define DIN  64
#define DM   256
#define DI   512
#define NN   32
#define RR   16
#define KK   4
#define MROWS (BB * LL)        // 16384 token rows

typedef __attribute__((ext_vector_type(2))) float v2f;
typedef __attribute__((ext_vector_type(8))) float v8f;

// ---------------------------------------------------------------------------
// Generic fp32 GEMM using V_WMMA_F32_16X16X4_F32.
//   C[M x N] = A[M x K](lda) @ W[K x N](ldb) (+ bias) (+ softplus)
// One wave computes a 64x16 tile (4 M-subtiles sharing each B fragment).
// mode: 0 = none, 1 = +bias, 2 = softplus(acc + bias)
// ---------------------------------------------------------------------------
__global__ void gemm_wmma_f32(const float* __restrict__ A, int lda,
                              const float* __restrict__ W, int ldb,
                              const float* __restrict__ bias,
                              float* __restrict__ C, int ldc,
                              int M, int N, int Kd, int mode) {
    const int wavesPerBlock = blockDim.x >> 5;
    const int wave = blockIdx.x * wavesPerBlock + (threadIdx.x >> 5);
    const int lane = threadIdx.x & 31;
    const int ntiles = N >> 4;          // 16-wide N tiles
    const int mtiles4 = M >> 6;         // 64-tall M tiles
    if (wave >= ntiles * mtiles4) return;

    const int tn = wave % ntiles;       // N tile index
    const int tm = wave / ntiles;       // 64-row M tile index
    const int half = lane >> 4;         // 0: K pair {0,1}, 1: K pair {2,3}
    const int r = lane & 15;            // row/col within tile

    v8f acc0 = {}; v8f acc1 = {}; v8f acc2 = {}; v8f acc3 = {};

    const int m0 = tm * 64;
    const int n0 = tn * 16;

    for (int k0 = 0; k0 < Kd; k0 += 4) {
        const int kk = k0 + 2 * half;
        // B fragment (4x16, K-major): lanes 0-15 -> K=kk row, cols n0+r
        v2f bf;
        bf.x = W[(size_t)kk * ldb + n0 + r];
        bf.y = W[(size_t)(kk + 1) * ldb + n0 + r];
        // A fragments: row = m0 + sub*16 + r, cols kk, kk+1 (contiguous -> b64)
        v2f a0 = *(const v2f*)(A + (size_t)(m0 +  0 + r) * lda + kk);
        v2f a1 = *(const v2f*)(A + (size_t)(m0 + 16 + r) * lda + kk);
        v2f a2 = *(const v2f*)(A + (size_t)(m0 + 32 + r) * lda + kk);
        v2f a3 = *(const v2f*)(A + (size_t)(m0 + 48 + r) * lda + kk);
        acc0 = __builtin_amdgcn_wmma_f32_16x16x4_f32(false, a0, false, bf, (short)0, acc0, false, false);
        acc1 = __builtin_amdgcn_wmma_f32_16x16x4_f32(false, a1, false, bf, (short)0, acc1, false, false);
        acc2 = __builtin_amdgcn_wmma_f32_16x16x4_f32(false, a2, false, bf, (short)0, acc2, false, false);
        acc3 = __builtin_amdgcn_wmma_f32_16x16x4_f32(false, a3, false, bf, (short)0, acc3, false, false);
    }

    const float bv = (mode >= 1) ? bias[n0 + r] : 0.0f;
    v8f accs[4] = {acc0, acc1, acc2, acc3};
    #pragma unroll
    for (int sub = 0; sub < 4; ++sub) {
        #pragma unroll
        for (int v = 0; v < 8; ++v) {
            // C/D layout: VGPR v holds row M = v + 8*half, lane r holds col N = r
            float val = accs[sub][v] + bv;
            if (mode == 2) {  // softplus
                val = (val > 20.0f) ? val : log1pf(__expf(val));
            }
            C[(size_t)(m0 + sub * 16 + v + 8 * half) * ldc + n0 + r] = val;
        }
    }
}

// ---------------------------------------------------------------------------
// Depthwise causal conv1d (K=4) + bias + SiLU.
// xr is [MROWS x 2*DI]; xc = xr[:, :DI]. Output xc_act [MROWS x DI].
// ---------------------------------------------------------------------------
__global__ void conv_silu_kernel(const float* __restrict__ xr,
                                 const float* __restrict__ w,   // [DI x K]
                                 const float* __restrict__ cb,  // [DI]
                                 float* __restrict__ out) {
    const int idx = blockIdx.x * blockDim.x + threadIdx.x;
    if (idx >= MROWS * DI) return;
    const int row = idx / DI;           // b*LL + t
    const int d = idx - row * DI;
    const int b = row / LL;
    const int t = row - b * LL;
    float s = cb[d];
    #pragma unroll
    for (int k = 0; k < KK; ++k) {
        const int tt = t - (KK - 1) + k;
        if (tt >= 0) {
            s = fmaf(w[d * KK + k], xr[(size_t)(b * LL + tt) * (2 * DI) + d], s);
        }
    }
    // SiLU
    out[idx] = s * (1.0f / (1.0f + __expf(-s)));
}

// ---------------------------------------------------------------------------
// Selective scan. One wave per (b, d); lane = state index n (wave32).
//   h[n] <- exp(dt*A[n]) * h[n] + (dt*u) * B[n];   y = sum_n h[n]*C[n]
// B/C live in xdbl rows at offsets RR and RR+NN (row stride RR+2*NN = 80).
// ---------------------------------------------------------------------------
__global__ void scan_kernel(const float* __restrict__ dt,     // [MROWS x DI]
                            const float* __restrict__ u,      // [MROWS x DI]
                            const float* __restrict__ xdbl,   // [MROWS x 80]
                            const float* __restrict__ A_log,  // [DI x NN]
                            float* __restrict__ y) {          // [MROWS x DI]
    const int wid = blockIdx.x * (blockDim.x >> 5) + (threadIdx.x >> 5);
    const int lane = threadIdx.x & 31;
    const int b = wid >> 9;             // / DI
    const int d = wid & (DI - 1);
    const float An = -__expf(A_log[d * NN + lane]);   // A = -exp(A_log)
    float h = 0.0f;
    const size_t rowBase = (size_t)b * LL;
    for (int t = 0; t < LL; ++t) {
        const size_t row = rowBase + t;
        const float dtv = dt[row * DI + d];
        const float uv  = u[row * DI + d];
        const float Bv  = xdbl[row * (RR + 2 * NN) + RR + lane];
        const float Cv  = xdbl[row * (RR + 2 * NN) + RR + NN + lane];
        const float a = __expf(dtv * An);
        h = fmaf(dtv * uv, Bv, a * h);
        float p = h * Cv;
        #pragma unroll
        for (int off = 16; off; off >>= 1) p += __shfl_xor(p, off, 32);
        if (lane == 0) y[row * DI + d] = p;
    }
}

// ---------------------------------------------------------------------------
// Gating epilogue: y = (y_scan + xc * Dp[d]) * silu(res), res = xr[:, DI+d]
// ---------------------------------------------------------------------------
__global__ void gate_kernel(float* __restrict__ y,
                            const float* __restrict__ xc,
                            const float* __restrict__ xr,
                            const float* __restrict__ Dp) {
    const int idx = blockIdx.x * blockDim.x + threadIdx.x;
    if (idx >= MROWS * DI) return;
    const int row = idx / DI;
    const int d = idx - row * DI;
    const float res = xr[(size_t)row * (2 * DI) + DI + d];
    const float sres = res * (1.0f / (1.0f + __expf(-res)));
    y[idx] = (y[idx] + xc[idx] * Dp[d]) * sres;
}

// ---------------------------------------------------------------------------
// Host-side orchestration
// ---------------------------------------------------------------------------
static inline void launch_gemm(const float* A, int lda, const float* W, int ldb,
                               const float* bias, float* C, int ldc,
                               int M, int N, int Kd, int mode, hipStream_t s) {
    const int tiles = (M / 64) * (N / 16);
    const int wavesPerBlock = 4;                     // 128 threads
    const int blocks = (tiles + wavesPerBlock - 1) / wavesPerBlock;
    gemm_wmma_f32<<<blocks, wavesPerBlock * 32, 0, s>>>(A, lda, W, ldb, bias, C, ldc, M, N, Kd, mode);
}

extern "C" void kernel_launch(void* const* d_in, const int* in_sizes, int n_in,
                              void* d_out, int out_size, void* d_ws, size_t ws_size,
                              hipStream_t stream) {
    const float* x        = (const float*)d_in[0];   // [B,L,DIN]
    const float* inp_w    = (const float*)d_in[1];   // [DIN,DM]
    const float* inp_b    = (const float*)d_in[2];   // [DM]
    const float* in_w     = (const float*)d_in[3];   // [2,DM,2*DI]
    const float* conv_w   = (const float*)d_in[4];   // [2,DI,1,K]
    const float* conv_b   = (const float*)d_in[5];   // [2,DI]
    const float* xproj_w  = (const float*)d_in[6];   // [2,DI,R+2N]
    const float* dtproj_w = (const float*)d_in[7];   // [2,R,DI]
    const float* dtproj_b = (const float*)d_in[8];   // [2,DI]
    const float* A_log    = (const float*)d_in[9];   // [2,DI,N]
    const float* Dp       = (const float*)d_in[10];  // [2,DI]
    const float* out_w    = (const float*)d_in[11];  // [2,DI,DM]
    const float* outp_w   = (const float*)d_in[12];  // [DM,DIN]
    const float* outp_b   = (const float*)d_in[13];  // [DIN]
    float* out = (float*)d_out;

    float* ws = (float*)d_ws;
    const size_t OFF_H    = 0;                               // MROWS*DM
    const size_t OFF_XR   = OFF_H    + (size_t)MROWS * DM;   // MROWS*2*DI
    const size_t OFF_XC   = OFF_XR   + (size_t)MROWS * 2 * DI;
    const size_t OFF_XDBL = OFF_XC   + (size_t)MROWS * DI;   // MROWS*80
    const size_t OFF_DT   = OFF_XDBL + (size_t)MROWS * (RR + 2 * NN);
    const size_t OFF_Y    = OFF_DT   + (size_t)MROWS * DI;
    float* hbuf  = ws + OFF_H;
    float* xrbuf = ws + OFF_XR;
    float* xcbuf = ws + OFF_XC;
    float* xdbl  = ws + OFF_XDBL;
    float* dtbuf = ws + OFF_DT;
    float* ybuf  = ws + OFF_Y;

    const int M = MROWS;
    const int ethreads = 256;
    const int eblocks = (M * DI + ethreads - 1) / ethreads;

    // h = x @ inp_w + inp_b
    launch_gemm(x, DIN, inp_w, DM, inp_b, hbuf, DM, M, DM, DIN, 1, stream);

    for (int i = 0; i < 2; ++i) {
        // xr = h @ in_w[i]                         [M x 1024]
        launch_gemm(hbuf, DM, in_w + (size_t)i * DM * 2 * DI, 2 * DI,
                    nullptr, xrbuf, 2 * DI, M, 2 * DI, DM, 0, stream);
        // xc = silu(causal_depthwise_conv(xc) + conv_b)
        conv_silu_kernel<<<eblocks, ethreads, 0, stream>>>(
            xrbuf, conv_w + (size_t)i * DI * KK, conv_b + (size_t)i * DI, xcbuf);
        // xdbl = xc @ xproj_w[i]                   [M x 80]
        launch_gemm(xcbuf, DI, xproj_w + (size_t)i * DI * (RR + 2 * NN), RR + 2 * NN,
                    nullptr, xdbl, RR + 2 * NN, M, RR + 2 * NN, DI, 0, stream);
        // delta = softplus(xdbl[:, :R] @ dtproj_w[i] + dtproj_b[i])  [M x 512]
        launch_gemm(xdbl, RR + 2 * NN, dtproj_w + (size_t)i * RR * DI, DI,
                    dtproj_b + (size_t)i * DI, dtbuf, DI, M, DI, RR, 2, stream);
        // selective scan: 4096 waves (one per (b,d)), 8 waves/block
        scan_kernel<<<(BB * DI) / 8, 256, 0, stream>>>(
            dtbuf, xcbuf, xdbl, A_log + (size_t)i * DI * NN, ybuf);
        // y = (y + xc*Dp) * silu(res)
        gate_kernel<<<eblocks, ethreads, 0, stream>>>(ybuf, xcbuf, xrbuf,
                                                      Dp + (size_t)i * DI);
        // h = y @ out_w[i]                         [M x 256]
        launch_gemm(ybuf, DI, out_w + (size_t)i * DI * DM, DM,
                    nullptr, hbuf, DM, M, DM, DI, 0, stream);
    }

    // x_rec = h @ outp_w + outp_b                  [M x 64]
    launch_gemm(hbuf, DM, outp_w, DIN, outp_b, out, DIN, M, DIN, DM, 1, stream);
}